// CrossEgoContrastive_27685359190725
// MI455X (gfx1250) — compile-verified
//
#include <hip/hip_runtime.h>
#include <hip/hip_bf16.h>

// CDNA5 / gfx1250, wave32. FP32 end-to-end using native V_WMMA_F32_16X16X4_F32.

typedef float v2f __attribute__((ext_vector_type(2)));
typedef float v8f __attribute__((ext_vector_type(8)));

#define NROWS  4096   // B*A
#define EMBEDD 1024
#define HIDDEN 1024
#define PROJ   256

__device__ __forceinline__ v8f wmma_f32_16x16x4(v2f a, v2f b, v8f c) {
  // 8 args: (neg_a, A, neg_b, B, c_mod, C, reuse_a, reuse_b)
  return __builtin_amdgcn_wmma_f32_16x16x4_f32(false, a, false, b, (short)0, c,
                                               false, false);
}

__device__ __forceinline__ float gelu_exact(float v) {
  return 0.5f * v * (1.0f + erff(v * 0.70710678118654752440f));
}

// softplus(u) = log(1 + e^u), numerically stable
__device__ __forceinline__ float softplus_stable(float u) {
  return fmaxf(u, 0.0f) + log1pf(expf(-fabsf(u)));
}

// C[M,NC] = act(A[M,K] @ B[K,NC] + bias)
// Block: 256 threads = 8 waves, each wave a 16x64 strip -> 128x64 block tile.
// WMMA f32 16x16x4 fragment layout (ISA 7.12.2):
//   A 16x4 : lane l<16 -> row=l, vgpr{0,1}=K{0,1};  lane>=16 -> K{2,3}
//   B 4x16 : lane l<16 -> col=l, vgpr{0,1}=K{0,1};  lane>=16 -> K{2,3}
//   C 16x16: vgpr r, lane l<16 -> (M=r,     N=l)
//                    lane l>=16 -> (M=r+8,  N=l-16)
template <bool APPLY_GELU>
__global__ __launch_bounds__(256) void gemm_bias_act(
    const float* __restrict__ A, const float* __restrict__ B,
    const float* __restrict__ bias, float* __restrict__ C, int K, int NC) {
  const int wave = threadIdx.x >> 5;
  const int lane = threadIdx.x & 31;
  const int half = lane >> 4;  // 0: K pair {0,1}, 1: K pair {2,3}
  const int l16  = lane & 15;
  const int row0 = blockIdx.x * 128 + wave * 16;
  const int col0 = blockIdx.y * 64;

  v8f acc[4];
#pragma unroll
  for (int t = 0; t < 4; ++t)
#pragma unroll
    for (int r = 0; r < 8; ++r) acc[t][r] = 0.0f;

  // per-lane A row pointer: contiguous pair -> global_load_b64
  const float* ap = A + (size_t)(row0 + l16) * K + 2 * half;

#pragma unroll 4
  for (int k = 0; k < K; k += 4) {
    v2f a;
    a.x = ap[k + 0];
    a.y = ap[k + 1];
    const float* bp = B + (size_t)(k + 2 * half) * NC + col0 + l16;
#pragma unroll
    for (int t = 0; t < 4; ++t) {
      v2f b;
      b.x = bp[t * 16];
      b.y = bp[t * 16 + NC];
      acc[t] = wmma_f32_16x16x4(a, b, acc[t]);
    }
  }

#pragma unroll
  for (int t = 0; t < 4; ++t) {
    const int col = col0 + t * 16 + l16;
    const float bv = bias[col];
#pragma unroll
    for (int r = 0; r < 8; ++r) {
      const int row = row0 + r + 8 * half;
      float v = acc[t][r] + bv;
      if (APPLY_GELU) v = gelu_exact(v);
      C[(size_t)row * NC + col] = v;
    }
  }
}

// z[row,:] = p[row,:] / ||p[row,:]||  (PROJ == blockDim == 256)
__global__ __launch_bounds__(256) void row_normalize(
    const float* __restrict__ p, float* __restrict__ z) {
  __shared__ float red[256];
  const int row = blockIdx.x;
  const int tid = threadIdx.x;
  const float v = p[(size_t)row * PROJ + tid];
  red[tid] = v * v;
  __syncthreads();
#pragma unroll
  for (int s = 128; s > 0; s >>= 1) {
    if (tid < s) red[tid] += red[tid + s];
    __syncthreads();
  }
  const float n = sqrtf(red[0]);
  z[(size_t)row * PROJ + tid] = v / n;
}

// logits = (z @ z^T)*scale + bias; per-block deterministic loss partial sums.
__global__ __launch_bounds__(256) void logits_loss(
    const float* __restrict__ z, const float* __restrict__ log_scale,
    const float* __restrict__ logit_bias, float* __restrict__ logits,
    float* __restrict__ partials) {
  const int wave = threadIdx.x >> 5;
  const int lane = threadIdx.x & 31;
  const int half = lane >> 4;
  const int l16  = lane & 15;
  const int row0 = blockIdx.x * 128 + wave * 16;
  const int col0 = blockIdx.y * 64;
  const float scale = expf(log_scale[0]);
  const float lb = logit_bias[0];

  v8f acc[4];
#pragma unroll
  for (int t = 0; t < 4; ++t)
#pragma unroll
    for (int r = 0; r < 8; ++r) acc[t][r] = 0.0f;

  const float* ap = z + (size_t)(row0 + l16) * PROJ + 2 * half;

#pragma unroll 4
  for (int k = 0; k < PROJ; k += 4) {
    v2f a;
    a.x = ap[k + 0];
    a.y = ap[k + 1];
#pragma unroll
    for (int t = 0; t < 4; ++t) {
      // B = z^T : B[k][n] = z[n][k]; contiguous pair per lane -> b64
      const float* bp = z + (size_t)(col0 + t * 16 + l16) * PROJ + k + 2 * half;
      v2f b;
      b.x = bp[0];
      b.y = bp[1];
      acc[t] = wmma_f32_16x16x4(a, b, acc[t]);
    }
  }

  float lsum = 0.0f;
#pragma unroll
  for (int t = 0; t < 4; ++t) {
    const int col = col0 + t * 16 + l16;
    const int cg = col >> 4;  // col / A(=16)
#pragma unroll
    for (int r = 0; r < 8; ++r) {
      const int row = row0 + r + 8 * half;
      const float lg = acc[t][r] * scale + lb;
      logits[(size_t)row * NROWS + col] = lg;
      const float m = ((row >> 4) == cg) ? 1.0f : -1.0f;
      lsum += softplus_stable(-m * lg);  // = -log_sigmoid(m*lg)
    }
  }

  __shared__ float red[256];
  red[threadIdx.x] = lsum;
  __syncthreads();
#pragma unroll
  for (int s = 128; s > 0; s >>= 1) {
    if (threadIdx.x < s) red[threadIdx.x] += red[threadIdx.x + s];
    __syncthreads();
  }
  if (threadIdx.x == 0)
    partials[blockIdx.y * gridDim.x + blockIdx.x] = red[0];
}

// Deterministic fixed-order final reduction of per-block partials.
__global__ __launch_bounds__(256) void loss_final(
    const float* __restrict__ partials, float* __restrict__ loss, int n) {
  __shared__ float red[256];
  float s = 0.0f;
  for (int i = threadIdx.x; i < n; i += 256) s += partials[i];
  red[threadIdx.x] = s;
  __syncthreads();
#pragma unroll
  for (int t = 128; t > 0; t >>= 1) {
    if (threadIdx.x < t) red[threadIdx.x] += red[threadIdx.x + t];
    __syncthreads();
  }
  if (threadIdx.x == 0) loss[0] = red[0] * (1.0f / (float)NROWS);
}

extern "C" void kernel_launch(void* const* d_in, const int* in_sizes, int n_in,
                              void* d_out, int out_size, void* d_ws,
                              size_t ws_size, hipStream_t stream) {
  const float* x  = (const float*)d_in[0];  // [4096,1024]
  const float* w1 = (const float*)d_in[1];  // [1024,1024]
  const float* b1 = (const float*)d_in[2];  // [1024]
  const float* w2 = (const float*)d_in[3];  // [1024,256]
  const float* b2 = (const float*)d_in[4];  // [256]
  const float* ls = (const float*)d_in[5];  // scalar log_scale
  const float* lb = (const float*)d_in[6];  // scalar logit_bias

  float* out    = (float*)d_out;
  float* emb    = out;                          // 4096*256 (z, reshaped)
  float* loss   = out + (size_t)NROWS * PROJ;   // 1
  float* logits = loss + 1;                     // 4096*4096

  float* h        = (float*)d_ws;               // 4096*1024 f32 (16 MB)
  float* p        = h + (size_t)NROWS * HIDDEN; // 4096*256  f32 (4 MB)
  float* partials = p + (size_t)NROWS * PROJ;   // 2048 f32

  // 1) h = gelu(x @ w1 + b1)
  gemm_bias_act<true><<<dim3(NROWS / 128, HIDDEN / 64), 256, 0, stream>>>(
      x, w1, b1, h, EMBEDD, HIDDEN);
  // 2) p = h @ w2 + b2
  gemm_bias_act<false><<<dim3(NROWS / 128, PROJ / 64), 256, 0, stream>>>(
      h, w2, b2, p, HIDDEN, PROJ);
  // 3) z = rownorm(p) -> embeddings output
  row_normalize<<<NROWS, 256, 0, stream>>>(p, emb);
  // 4) logits + per-block loss partials
  logits_loss<<<dim3(NROWS / 128, NROWS / 64), 256, 0, stream>>>(
      emb, ls, lb, logits, partials);
  // 5) deterministic loss reduce
  loss_final<<<1, 256, 0, stream>>>(partials, loss,
                                    (NROWS / 128) * (NROWS / 64));
}